// MSE_and_Contrastive_loss_71966472012556
// MI455X (gfx1250) — compile-verified
//
#include <hip/hip_runtime.h>

typedef float v2f __attribute__((ext_vector_type(2)));
typedef float v4f __attribute__((ext_vector_type(4)));
typedef float v8f __attribute__((ext_vector_type(8)));

// Workspace layout (floats):
#define WS_RECON 0      // [0..1]   final recon sums (unused slot; partials reduced in final)
#define WS_SQ    2      // [2..129] sq1 (64) then sq2 (64)
#define WS_CROSS 130    // [130..4225] 64x64 cross matrix
#define WS_PART  4352   // [4352..6399] recon partials: 1024 blocks x 2
#define NBLK_RECON 1024
#define LSTR 68         // padded LDS row stride (floats): 16B-aligned, conflict-free

// ---------------- Kernel 0: zero the cross accumulator ----------------
__global__ void init_cross_kernel(float* __restrict__ cross) {
    int t = blockIdx.x * blockDim.x + threadIdx.x;
    for (int i = t; i < 64 * 64; i += gridDim.x * blockDim.x) cross[i] = 0.0f;
}

// ---------------- Kernel 1: MSE partial sums (two arrays at once) ----------------
__global__ __launch_bounds__(256)
void recon_partial_kernel(const float* __restrict__ o1, const float* __restrict__ i1,
                          const float* __restrict__ o2, const float* __restrict__ i2,
                          float* __restrict__ part, int n4) {
    __shared__ float sm1[256];
    __shared__ float sm2[256];
    int t = threadIdx.x;
    long stride = (long)gridDim.x * 256;
    float s1 = 0.0f, s2 = 0.0f;
    for (long idx = (long)blockIdx.x * 256 + t; idx < n4; idx += stride) {
        v4f a = ((const v4f*)o1)[idx];
        v4f b = ((const v4f*)i1)[idx];
        v4f c = ((const v4f*)o2)[idx];
        v4f d = ((const v4f*)i2)[idx];
        v4f e1 = a - b;
        v4f e2 = c - d;
        s1 += e1.x * e1.x + e1.y * e1.y + e1.z * e1.z + e1.w * e1.w;
        s2 += e2.x * e2.x + e2.y * e2.y + e2.z * e2.z + e2.w * e2.w;
    }
    sm1[t] = s1; sm2[t] = s2;
    __syncthreads();
    for (int off = 128; off > 0; off >>= 1) {
        if (t < off) { sm1[t] += sm1[t + off]; sm2[t] += sm2[t + off]; }
        __syncthreads();
    }
    if (t == 0) {
        part[2 * blockIdx.x + 0] = sm1[0];
        part[2 * blockIdx.x + 1] = sm2[0];
    }
}

// ---------------- Kernel 2: per-row sum of squares for z1 / z2 ----------------
__global__ __launch_bounds__(256)
void sq_kernel(const float* __restrict__ z1, const float* __restrict__ z2,
               float* __restrict__ sq, int D) {
    __shared__ float sm[256];
    int b = blockIdx.x, t = threadIdx.x;
    const float* row = (b < 64) ? (z1 + (long)b * D) : (z2 + (long)(b - 64) * D);
    int n4 = D >> 2;
    float s = 0.0f;
    for (int j = t; j < n4; j += 256) {
        v4f v = ((const v4f*)row)[j];
        s += v.x * v.x + v.y * v.y + v.z * v.z + v.w * v.w;
    }
    sm[t] = s;
    __syncthreads();
    for (int off = 128; off > 0; off >>= 1) {
        if (t < off) sm[t] += sm[t + off];
        __syncthreads();
    }
    if (t == 0) sq[b] = sm[0];   // b<64 -> sq1[b]; b>=64 -> sq2[b-64] (contiguous)
}

// ---------------- Kernel 3: cross = z1f @ z2f^T via V_WMMA_F32_16X16X4_F32 ----------------
// 16 workgroups split the K dimension; each WG has 16 waves, one 16x16 tile per wave.
__global__ __launch_bounds__(512)
void cross_wmma_kernel(const float* __restrict__ z1, const float* __restrict__ z2,
                       float* __restrict__ cross, int D, int kPerBlock) {
    __shared__ float lz1[64 * LSTR];
    __shared__ float lz2[64 * LSTR];
    int t = threadIdx.x;
    int wave = t >> 5, lane = t & 31;
    int mi = wave >> 2, ni = wave & 3;            // tile coords in 4x4 tile grid
    int rA = mi * 16 + (lane & 15);               // A fragment: M = lane%16
    int rB = ni * 16 + (lane & 15);               // B fragment: N = lane%16
    int kc = (lane >> 4) * 2;                     // K sub-offset: lanes 16-31 hold K+2,K+3
    int kBase = blockIdx.x * kPerBlock;

    v8f acc = {};
    for (int k0 = kBase; k0 < kBase + kPerBlock; k0 += 64) {
        __syncthreads();
        // Cooperative stage of 64x64 f32 tiles of z1/z2 into padded LDS.
#pragma unroll
        for (int i = 0; i < 2; ++i) {
            int f4  = t + i * 512;                // 1024 float4 per array per chunk
            int row = f4 >> 4;
            int cv  = (f4 & 15) * 4;
            v4f a = *(const v4f*)(z1 + (long)row * D + k0 + cv);
            v4f b = *(const v4f*)(z2 + (long)row * D + k0 + cv);
            *(v4f*)(&lz1[row * LSTR + cv]) = a;   // 16B-aligned ds_store_b128
            *(v4f*)(&lz2[row * LSTR + cv]) = b;
        }
        __syncthreads();
#pragma unroll
        for (int kk = 0; kk < 64; kk += 4) {
            v2f a = *(const v2f*)(&lz1[rA * LSTR + kk + kc]);  // ds_load_b64
            v2f b = *(const v2f*)(&lz2[rB * LSTR + kk + kc]);
            acc = __builtin_amdgcn_wmma_f32_16x16x4_f32(
                false, a, false, b, (short)0, acc, false, false);
        }
    }
    // D layout: lane l, VGPR v -> row v + 8*(l/16), col l%16 (within 16x16 tile).
#pragma unroll
    for (int v = 0; v < 8; ++v) {
        int row = mi * 16 + v + ((lane >> 4) << 3);
        int col = ni * 16 + (lane & 15);
        atomicAdd(&cross[row * 64 + col], acc[v]);
    }
}

// ---------------- Kernel 4: assemble scalar loss ----------------
__global__ __launch_bounds__(256)
void final_kernel(const float* __restrict__ ws, float* __restrict__ out,
                  float nElems, int D) {
    __shared__ float sp1[256], sp2[256], sLP[256], sLN[256];
    int t = threadIdx.x;
    const float* part  = ws + WS_PART;
    const float* sq1   = ws + WS_SQ;
    const float* sq2   = ws + WS_SQ + 64;
    const float* cross = ws + WS_CROSS;

    float p1 = 0.0f, p2 = 0.0f;
#pragma unroll
    for (int i = 0; i < NBLK_RECON / 256; ++i) {
        int idx = t + i * 256;
        p1 += part[2 * idx + 0];
        p2 += part[2 * idx + 1];
    }

    float invD = 1.0f / (float)D;
    float lp = 0.0f, ln = 0.0f;
#pragma unroll
    for (int i = 0; i < 16; ++i) {
        int p = t * 16 + i;          // 4096 (m,n) pairs
        int m = p >> 6, n = p & 63;
        float d = (sq1[m] + sq2[n] - 2.0f * cross[p]) * invD;
        if (m == n) { if (d > 0.01f) lp += d - 0.01f; }
        else        { if (d < 0.4f)  ln += 0.4f - d; }
    }

    sp1[t] = p1; sp2[t] = p2; sLP[t] = lp; sLN[t] = ln;
    __syncthreads();
    for (int off = 128; off > 0; off >>= 1) {
        if (t < off) {
            sp1[t] += sp1[t + off]; sp2[t] += sp2[t + off];
            sLP[t] += sLP[t + off]; sLN[t] += sLN[t + off];
        }
        __syncthreads();
    }
    if (t == 0) {
        float recon = sp1[0] / nElems + sp2[0] / nElems;
        float lossP = 1.5f * (sLP[0] / 64.0f);
        float lossN = 0.5f * (sLN[0] / (64.0f * 63.0f));
        out[0] = recon + lossP + lossN;
    }
}

extern "C" void kernel_launch(void* const* d_in, const int* in_sizes, int n_in,
                              void* d_out, int out_size, void* d_ws, size_t ws_size,
                              hipStream_t stream) {
    const float* o1 = (const float*)d_in[0];
    const float* z1 = (const float*)d_in[1];
    const float* o2 = (const float*)d_in[2];
    const float* z2 = (const float*)d_in[3];
    const float* i1 = (const float*)d_in[4];
    const float* i2 = (const float*)d_in[5];
    float* ws  = (float*)d_ws;
    float* out = (float*)d_out;

    int n  = in_sizes[0];        // 64*3*256*256
    int n4 = n / 4;
    int D  = in_sizes[1] / 64;   // 16384

    const int kSplit = 16;
    int kPerBlock = D / kSplit;

    init_cross_kernel<<<4, 256, 0, stream>>>(ws + WS_CROSS);
    recon_partial_kernel<<<NBLK_RECON, 256, 0, stream>>>(o1, i1, o2, i2, ws + WS_PART, n4);
    sq_kernel<<<128, 256, 0, stream>>>(z1, z2, ws + WS_SQ, D);
    cross_wmma_kernel<<<kSplit, 512, 0, stream>>>(z1, z2, ws + WS_CROSS, D, kPerBlock);
    final_kernel<<<1, 256, 0, stream>>>(ws, out, (float)n, D);
}